// FeatureGeneratorMoe_6571299962924
// MI455X (gfx1250) — compile-verified
//
#include <hip/hip_runtime.h>
#include <math.h>

typedef __attribute__((ext_vector_type(2))) float v2f;
typedef __attribute__((ext_vector_type(8))) float v8f;

#define HID   16
#define NMLP  8

// ---- Branchless native-TRANS helpers (v_exp_f32 / v_log_f32 / v_rcp_f32) ----
__device__ __forceinline__ float frcp(float v)  { return __builtin_amdgcn_rcpf(v); }
__device__ __forceinline__ float fexp(float v)  { return __expf(v); }   // v_exp_f32
__device__ __forceinline__ float flog(float v)  { return __logf(v); }   // v_log_f32

__device__ __forceinline__ float tanh_fast(float v) {
    float t = fexp(-2.0f * fabsf(v));            // in (0, 1]
    float r = (1.0f - t) * frcp(1.0f + t);       // saturates to 1 for large |v|
    return copysignf(r, v);
}

__device__ __forceinline__ float erf_fast(float v) {
    // Abramowitz–Stegun 7.1.26, max abs err ~1.5e-7, branchless.
    float a  = fabsf(v);
    float t  = frcp(fmaf(0.3275911f, a, 1.0f));
    float p  = fmaf(1.061405429f, t, -1.453152027f);
    p = fmaf(p, t, 1.421413741f);
    p = fmaf(p, t, -0.284496736f);
    p = fmaf(p, t, 0.254829592f);
    float r = 1.0f - p * t * fexp(-a * a);
    return copysignf(r, v);
}

// Activation i applied to pre-activation v (matches reference _ACTS order).
__device__ __forceinline__ float moe_act(int i, float v) {
    switch (i) {
        case 0: return fmaxf(v, 0.0f);                               // relu
        case 1: return tanh_fast(v);                                 // tanh
        case 2: return v > 0.0f ? v : (fexp(v) - 1.0f);              // elu (select)
        case 3: return v * frcp(1.0f + fexp(-v));                    // silu
        case 4: return 0.5f * v * (1.0f + erf_fast(v * 0.7071067811865476f)); // gelu exact
        case 5: return v;                                            // none
        case 6: {                                                    // selu (select)
            const float a = 1.6732632423543772f, s = 1.0507009873554805f;
            return s * (v > 0.0f ? v : a * (fexp(v) - 1.0f));
        }
        default:                                                     // softplus, branchless
            return fmaxf(v, 0.0f) + flog(1.0f + fexp(-fabsf(v)));
    }
}

// One wave processes 16 points per tile with V_WMMA_F32_16X16X4_F32:
//   D[h, p] = sum_k A[h,k] * B[k,p],  A = [W1; b1]^T (16x4),  B = [re; im; amp; 1] (4x16)
__global__ void __launch_bounds__(256)
moe_wmma_kernel(const float* __restrict__ x,      // (N, 2)
                const float* __restrict__ W1,     // (8, 3, 16)
                const float* __restrict__ b1,     // (8, 16)
                const float* __restrict__ W2,     // (8, 16, 1)
                const float* __restrict__ b2,     // (8, 1)
                const float* __restrict__ ew,     // (12,)
                float* __restrict__ out,          // (N,)
                int npoints)
{
    const int lane = threadIdx.x & 31;
    const int half = lane >> 4;        // 0: lanes 0-15 (K=0,1 / hid 0-7), 1: lanes 16-31
    const int l16  = lane & 15;
    const int wave_in_blk  = threadIdx.x >> 5;
    const int waves_per_blk = blockDim.x >> 5;
    const int gwave  = blockIdx.x * waves_per_blk + wave_in_blk;
    const int nwaves = gridDim.x * waves_per_blk;

    // ---- Preload loop-invariant weights into registers (fully unrolled) ----
    // A-matrix ISA layout (16x4 f32): lane l<16 holds A[l, K=0..1] in VGPR0..1,
    // lane l+16 holds A[l, K=2..3].  Row M = hidden unit, K = {re, im, amp, bias}.
    v2f   Areg[NMLP];
    float w2r[NMLP][8];                // W2 slice for this half, scaled by expert weight
    float b2sum = 0.0f;
    #pragma unroll
    for (int i = 0; i < NMLP; ++i) {
        float a0 = half ? W1[(i * 3 + 2) * HID + l16]   // K=2: amp weight
                        : W1[(i * 3 + 0) * HID + l16];  // K=0: re  weight
        float a1 = half ? b1[i * HID + l16]             // K=3: bias
                        : W1[(i * 3 + 1) * HID + l16];  // K=1: im  weight
        Areg[i].x = a0;
        Areg[i].y = a1;
        const float w = ew[i];
        #pragma unroll
        for (int r = 0; r < 8; ++r)
            w2r[i][r] = W2[i * HID + half * 8 + r] * w;
        b2sum += b2[i] * w;
    }
    const float w_abs = ew[8], w_mod = ew[9], w_sin = ew[10], w_cos = ew[11];

    // ---- Grid-stride over 16-point tiles ----
    for (int tile = gwave; tile * 16 < npoints; tile += nwaves) {
        const int p = tile * 16 + l16;

        // Prefetch next tile's input (global_prefetch_b8; speculative, OOB-safe).
        __builtin_prefetch(&x[2 * (p + nwaves * 16)], 0, 3);

        const float re  = x[2 * p + 0];
        const float im  = x[2 * p + 1];
        const float amp = __builtin_amdgcn_sqrtf(fmaf(re, re, im * im)); // native v_sqrt_f32

        // B-matrix (4x16 f32): lanes 0-15 hold rows K=0,1 (re, im) in VGPR0,1;
        // lanes 16-31 hold rows K=2,3 (amp, 1.0).
        v2f Breg;
        Breg.x = half ? amp  : re;
        Breg.y = half ? 1.0f : im;

        // Accumulate weighted layer-2 dot products. b2 terms added once (lower half).
        float acc = half ? 0.0f : b2sum;

        #pragma unroll
        for (int i = 0; i < NMLP; ++i) {
            v8f c = {};
            // D[h, p]: lane p -> hidden 0-7 in c[0..7]; lane p+16 -> hidden 8-15.
            c = __builtin_amdgcn_wmma_f32_16x16x4_f32(
                    /*neg_a=*/false, Areg[i],
                    /*neg_b=*/false, Breg,
                    /*c_mod=*/(short)0, c,
                    /*reuse_a=*/false, /*reuse_b=*/false);
            #pragma unroll
            for (int r = 0; r < 8; ++r)
                acc = fmaf(moe_act(i, c[r]), w2r[i][r], acc);
        }

        // Combine hidden halves: lane p + lane p+16 -> full MLP contribution.
        acc += __shfl_xor(acc, 16, 32);

        // Analytic experts (identical in both halves).
        float s, co;
        __sincosf(amp, &s, &co);
        acc = fmaf(w_abs, fabsf(re) + fabsf(im), acc);
        acc = fmaf(w_mod, amp, acc);
        acc = fmaf(w_sin, s,  acc);
        acc = fmaf(w_cos, co, acc);

        if (!half)
            out[p] = acc;
    }
}

extern "C" void kernel_launch(void* const* d_in, const int* in_sizes, int n_in,
                              void* d_out, int out_size, void* d_ws, size_t ws_size,
                              hipStream_t stream) {
    const float* x  = (const float*)d_in[0];
    const float* W1 = (const float*)d_in[1];
    const float* b1 = (const float*)d_in[2];
    const float* W2 = (const float*)d_in[3];
    const float* b2 = (const float*)d_in[4];
    const float* ew = (const float*)d_in[5];
    float* out = (float*)d_out;

    const int npoints = in_sizes[0] / 2;               // x is (N, 2)
    const int threads = 256;                            // 8 waves/block (wave32)
    const int waves_per_blk = threads / 32;
    const int tiles = (npoints + 15) / 16;
    int blocks = (tiles + waves_per_blk - 1) / waves_per_blk;
    if (blocks > 4096) blocks = 4096;                   // grid-stride covers the rest
    if (blocks < 1) blocks = 1;

    moe_wmma_kernel<<<blocks, threads, 0, stream>>>(x, W1, b1, W2, b2, ew, out, npoints);
}